// VectorQuantizer_86784109183322
// MI455X (gfx1250) — compile-verified
//
#include <hip/hip_runtime.h>
#include <hip/hip_bf16.h>
#include <math.h>

// ---------------------------------------------------------------------------
// VQ-VAE vector quantizer for MI455X (gfx1250, wave32, WMMA)
//   z:     [16, 256, 32, 32] f32   -> zf [N=16384, D=256]
//   emb_w: [8192, 256] f32
// Outputs (flat f32 concat):
//   z_q_out [16,256,32,32] | loss | perplexity | encodings [16384,8192] | indices [16384]
// Strategy:
//   - distance GEMM on V_WMMA_F32_16X16X4_F32 (exact fp32 matrix path)
//   - K split in 2 halves across blocks (256 WGs) -> per-row partial argmin in ws
//   - synchronous LDS staging + global_prefetch_b8 of the next chunk
//     (no persistent staging regs -> no scratch spills)
//   - post kernel: merge halves, gather z_q, one-hot, histogram, loss partials
// ---------------------------------------------------------------------------

typedef __attribute__((ext_vector_type(2))) float v2f;
typedef __attribute__((ext_vector_type(8))) float v8f;

#define NROWS   16384      // N
#define KCODES  8192       // codebook entries
#define KSPLIT  2
#define KHALF   (KCODES / KSPLIT)
#define DDIM    256        // embedding dim
#define CHUNK   32         // codebook columns staged in LDS per step (32KB)
#define BPAD    260        // padded LDS row stride (floats): bank-stride 4 -> conflict free

#define OUT_ZQ   0
#define OUT_LOSS 4194304
#define OUT_PERP 4194305
#define OUT_ENC  4194306
#define OUT_IDX  138412034ll

// ---------------------------------------------------------------------------
// Prep: e2[k] = |emb_w[k]|^2  (one wave per codebook row), zero histogram
// ---------------------------------------------------------------------------
__global__ __launch_bounds__(256) void vq_prep_kernel(
    const float* __restrict__ emb, float* __restrict__ e2, int* __restrict__ counts)
{
    int t = threadIdx.x;
    int wave = t >> 5, lane = t & 31;
    int k = blockIdx.x * 8 + wave;
    const float* row = emb + (size_t)k * DDIM;
    float s = 0.f;
#pragma unroll
    for (int i = 0; i < 8; ++i) {
        float v = row[lane + 32 * i];
        s = fmaf(v, v, s);
    }
#pragma unroll
    for (int off = 16; off >= 1; off >>= 1) s += __shfl_xor(s, off, 32);
    if (lane == 0) e2[k] = s;
    if (t < 8) counts[blockIdx.x * 8 + t] = 0;
}

// ---------------------------------------------------------------------------
// Zero the one-hot encodings region (float2 stores; 8B-aligned offset)
// ---------------------------------------------------------------------------
__global__ __launch_bounds__(256) void vq_zero_kernel(float2* __restrict__ enc)
{
    size_t i = (size_t)blockIdx.x * 256 + threadIdx.x;   // 67108864 float2 total
    enc[i] = make_float2(0.f, 0.f);
}

// ---------------------------------------------------------------------------
// Main: fp32 WMMA distance GEMM + per-half argmin.
// grid = 256: block (wg,half); 256 threads = 8 waves; each wave owns 16 rows
// and scans KHALF codes; next chunk is warmed with global_prefetch_b8.
// ---------------------------------------------------------------------------
__global__ __launch_bounds__(256) void vq_main_kernel(
    const float* __restrict__ z, const float* __restrict__ emb,
    const float* __restrict__ e2,
    float* __restrict__ wsval, int* __restrict__ wsidx)
{
    __shared__ float Bs[CHUNK * BPAD];

    const int t    = threadIdx.x;
    const int wave = t >> 5;
    const int lane = t & 31;
    const int lm   = lane & 15;   // M index (A/C) or N index (B/C)
    const int g    = lane >> 4;   // K-pair selector within WMMA operand

    const int half = blockIdx.x & 1;
    const int wg   = blockIdx.x >> 1;
    const int cb0  = half * KHALF;
    const int cb1  = cb0 + KHALF;

    // rows n0..n0+15 for this wave; n = b*1024 + h*32 + w (16 rows stay in one b,h)
    const int n0 = wg * 128 + wave * 16;
    const int bb = n0 >> 10;
    const int hh = (n0 >> 5) & 31;
    const int w0 = n0 & 31;       // 0 or 16

    // per-lane base into z for row (lm): element (bb, c=0, hh, w0+lm); c-stride = 1024
    const size_t zoff = (((size_t)bb * 256) * 32 + hh) * 32 + w0 + lm;
    const float* zrow = z + zoff;

    // A tile in registers, exact WMMA f32 16x16x4 layout: a[j] = A[lm][4j+2g .. +1]
    v2f a[64];
#pragma unroll
    for (int j = 0; j < 64; ++j) {
        const int d0 = 4 * j + 2 * g;
        a[j].x = zrow[(size_t)d0 * 1024];
        a[j].y = zrow[(size_t)(d0 + 1) * 1024];
    }

    float minv[8];
    int   mini[8];
#pragma unroll
    for (int r = 0; r < 8; ++r) { minv[r] = 3.402823466e+38f; mini[r] = cb0; }

    const int scol = t >> 6;          // 0..3
    const int sd4  = (t & 63) << 2;   // 0..252 step 4

    for (int cb = cb0; cb < cb1; cb += CHUNK) {
        __syncthreads();
        // cooperative stage of CHUNK codebook rows into LDS (transient regs only)
#pragma unroll
        for (int i = 0; i < 8; ++i) {
            const int c2 = scol + 4 * i;  // 0..31
            const float4 v = *(const float4*)(emb + (size_t)(cb + c2) * DDIM + sd4);
            *(float4*)(&Bs[c2 * BPAD + sd4]) = v;
        }
        __syncthreads();

        // warm the next 32KB chunk (256 x 128B lines; 1 line/thread) while we
        // compute: global_prefetch_b8 costs no VGPRs and no LOADcnt.
        if (cb + CHUNK < cb1) {
            const char* nxt = (const char*)(emb + (size_t)(cb + CHUNK) * DDIM);
            __builtin_prefetch(nxt + (size_t)t * 128, 0, 3);
        }

#pragma unroll
        for (int ct = 0; ct < CHUNK / 16; ++ct) {
            v8f c = {0.f, 0.f, 0.f, 0.f, 0.f, 0.f, 0.f, 0.f};
            const float* bp = &Bs[(ct * 16 + lm) * BPAD + 2 * g];
#pragma unroll
            for (int j = 0; j < 64; ++j) {
                const v2f b = *(const v2f*)(bp + 4 * j);
                // D = A(16x4,f32) * B(4x16,f32) + C ; accumulate z.e over D=256
                c = __builtin_amdgcn_wmma_f32_16x16x4_f32(
                        false, a[j], false, b, (short)0, c, false, false);
            }
            const int colb = cb + ct * 16 + lm;
            const float e2v = e2[colb];
#pragma unroll
            for (int r = 0; r < 8; ++r) {
                const float s = fmaf(-2.f, c[r], e2v);  // |e|^2 - 2 z.e
                if (s < minv[r]) { minv[r] = s; mini[r] = colb; }
            }
        }
    }

    // butterfly argmin within each 16-lane half (tie-break: lowest index == first min)
#pragma unroll
    for (int r = 0; r < 8; ++r) {
#pragma unroll
        for (int off = 1; off < 16; off <<= 1) {
            const float ov = __shfl_xor(minv[r], off, 32);
            const int   oi = __shfl_xor(mini[r], off, 32);
            if (ov < minv[r] || (ov == minv[r] && oi < mini[r])) { minv[r] = ov; mini[r] = oi; }
        }
    }

    // per-row partials for this K-half (lane 0: rows 0..7, lane 16: rows 8..15)
    if (lane == 0 || lane == 16) {
        const int base_r = (lane == 0) ? 0 : 8;
        float* wv = wsval + (size_t)half * NROWS;
        int*   wi = wsidx + (size_t)half * NROWS;
        for (int r = 0; r < 8; ++r) {
            wv[n0 + base_r + r] = minv[r];
            wi[n0 + base_r + r] = mini[r];
        }
    }
}

// ---------------------------------------------------------------------------
// Post: merge K-halves, write indices/one-hot/histogram, gather z_q (straight-
// through forward value), transposed write-back, per-block loss partials.
// grid = 128, 256 threads = 8 waves x 16 rows.
// ---------------------------------------------------------------------------
__global__ __launch_bounds__(256) void vq_post_kernel(
    const float* __restrict__ z, const float* __restrict__ emb,
    const float* __restrict__ wsval, const int* __restrict__ wsidx,
    float* __restrict__ out, int* __restrict__ counts, float* __restrict__ lossp)
{
    __shared__ float red[8];

    const int t    = threadIdx.x;
    const int wave = t >> 5;
    const int lane = t & 31;
    const int lm   = lane & 15;
    const int g    = lane >> 4;

    const int n0  = blockIdx.x * 128 + wave * 16;
    const int row = n0 + lm;
    const int bb  = n0 >> 10;
    const int hh  = (n0 >> 5) & 31;
    const int w0  = n0 & 31;

    // merge the two half-range argmins; half0 indices < half1 indices, so a
    // strict '<' keeps the first (lowest-index) minimum on exact ties.
    const float v0 = wsval[row];
    const float v1 = wsval[NROWS + row];
    const int   i0 = wsidx[row];
    const int   i1 = wsidx[NROWS + row];
    const int   id = (v1 < v0) ? i1 : i0;

    if (g == 0) {
        out[OUT_IDX + row] = (float)id;
        out[OUT_ENC + (size_t)row * KCODES + id] = 1.0f;
        atomicAdd(&counts[id], 1);
    }

    // gather codebook row, write z_q transposed back to [b,c,h,w], loss accum
    const size_t zoff = (((size_t)bb * 256) * 32 + hh) * 32 + w0 + lm;
    const float* zrow = z + zoff;
    float*       zq   = out + OUT_ZQ + zoff;
    const float* erow = emb + (size_t)id * DDIM;

    float acc = 0.f;
#pragma unroll
    for (int j = 0; j < 64; ++j) {
        const int c0 = 4 * j + 2 * g;
        const float e0 = erow[c0];
        const float e1 = erow[c0 + 1];
        const float z0 = zrow[(size_t)c0 * 1024];
        const float z1 = zrow[(size_t)(c0 + 1) * 1024];
        zq[(size_t)c0 * 1024]       = e0;
        zq[(size_t)(c0 + 1) * 1024] = e1;
        const float d0 = e0 - z0;
        const float d1 = e1 - z1;
        acc = fmaf(d0, d0, acc);
        acc = fmaf(d1, d1, acc);
    }
#pragma unroll
    for (int off = 16; off >= 1; off >>= 1) acc += __shfl_xor(acc, off, 32);
    if (lane == 0) red[wave] = acc;
    __syncthreads();
    if (t == 0) {
        float s = 0.f;
        for (int w = 0; w < 8; ++w) s += red[w];
        lossp[blockIdx.x] = s;   // deterministic per-block partial
    }
}

// ---------------------------------------------------------------------------
// Final: loss scalar + perplexity from histogram (single block, deterministic)
// ---------------------------------------------------------------------------
__global__ __launch_bounds__(256) void vq_final_kernel(
    const float* __restrict__ lossp, const int* __restrict__ counts,
    float* __restrict__ out)
{
    __shared__ float sh[256];
    const int t = threadIdx.x;
    float ps = 0.f;
    for (int k = t; k < KCODES; k += 256) {
        const float p = (float)counts[k] * (1.0f / (float)NROWS);
        ps += p * logf(p + 1e-10f);
    }
    sh[t] = ps;
    __syncthreads();
    for (int s = 128; s >= 1; s >>= 1) {
        if (t < s) sh[t] += sh[t + s];
        __syncthreads();
    }
    if (t == 0) {
        float ls = 0.f;
        for (int i = 0; i < 128; ++i) ls += lossp[i];
        out[OUT_LOSS] = 1.25f * ls / ((float)NROWS * (float)DDIM);  // (1+BETA)*mean
        out[OUT_PERP] = expf(-sh[0]);
    }
}

// ---------------------------------------------------------------------------
extern "C" void kernel_launch(void* const* d_in, const int* in_sizes, int n_in,
                              void* d_out, int out_size, void* d_ws, size_t ws_size,
                              hipStream_t stream) {
    const float* z   = (const float*)d_in[0];
    const float* emb = (const float*)d_in[1];
    float* out = (float*)d_out;

    // workspace layout (bytes): e2 | counts | lossp | wsval[2*N] | wsidx[2*N]
    float* e2     = (float*)d_ws;                                   // 8192 f32
    int*   counts = (int*)((char*)d_ws + 8192 * 4);                 // 8192 i32
    float* lossp  = (float*)((char*)d_ws + 16384 * 4);              // 128 f32
    float* wsval  = (float*)((char*)d_ws + 16512 * 4);              // 2*16384 f32
    int*   wsidx  = (int*)((char*)d_ws + (16512 + 2 * 16384) * 4);  // 2*16384 i32

    // 1) codebook norms + zero histogram
    vq_prep_kernel<<<dim3(KCODES / 8), dim3(256), 0, stream>>>(emb, e2, counts);
    // 2) zero the 537MB one-hot region (float2 stores: 67108864 elements)
    vq_zero_kernel<<<dim3(262144), dim3(256), 0, stream>>>((float2*)(out + OUT_ENC));
    // 3) WMMA distance GEMM + per-half argmin (256 blocks: 128 row-tiles x 2 K-halves)
    vq_main_kernel<<<dim3((NROWS / 128) * KSPLIT), dim3(256), 0, stream>>>(
        z, emb, e2, wsval, wsidx);
    // 4) merge + gather + one-hot + histogram + loss partials
    vq_post_kernel<<<dim3(NROWS / 128), dim3(256), 0, stream>>>(
        z, emb, wsval, wsidx, out, counts, lossp);
    // 5) scalars
    vq_final_kernel<<<dim3(1), dim3(256), 0, stream>>>(lossp, counts, out);
}